// SS2D_49185965474191
// MI455X (gfx1250) — compile-verified
//
#include <hip/hip_runtime.h>

#define BATCH 4
#define HH 64
#define WW 64
#define LL 4096          // HH*WW
#define DM 96
#define DE 192
#define KD 4
#define NS 16
#define RR 6
#define CDL 38           // RR + 2*NS
#define LN_EPS 1e-5f

typedef float v2f __attribute__((ext_vector_type(2)));
typedef float v8f __attribute__((ext_vector_type(8)));

__device__ __forceinline__ float gelu_tanh(float x) {
  const float c0 = 0.7978845608028654f;  // sqrt(2/pi)
  return 0.5f * x * (1.0f + tanhf(c0 * (x + 0.044715f * x * x * x)));
}

// CrossScan gather map; it is its own inverse, so CrossMerge scatters through it too.
__device__ __forceinline__ int scan_pos(int k, int l) {
  int ll = (k & 2) ? (LL - 1 - l) : l;
  if (k & 1) {                 // column-major direction: scan index = w*H + h
    int h = ll & (HH - 1);
    int w = ll >> 6;
    return h * WW + w;
  }
  return ll;
}

// ---------------- Kernel 1: in_proj (1x1 conv, 384x96 GEMM) + split + gelu(z) ---
// One wave per 16x16 output tile. A = in_proj_w (row-major o,c), B = x tokens.
__global__ __launch_bounds__(32)
void k_inproj(const float* __restrict__ x, const float* __restrict__ wproj,
              float* __restrict__ xh_raw, float* __restrict__ z_gelu) {
  const int lane = threadIdx.x;
  const int half = lane >> 4;
  const int lr   = lane & 15;
  const int m0   = blockIdx.x * 16;        // output-channel tile base (0..383)
  const int t    = blockIdx.y * 16 + lr;   // global token (b*LL + l)
  const int m    = m0 + lr;                // A row

  v8f acc = {};
  #pragma unroll 4
  for (int c0 = 0; c0 < DM; c0 += 4) {
    v2f a, b;
    int ks = c0 + 2 * half;
    a.x = wproj[m * DM + ks];
    a.y = wproj[m * DM + ks + 1];
    b.x = x[(size_t)t * DM + ks];
    b.y = x[(size_t)t * DM + ks + 1];
    acc = __builtin_amdgcn_wmma_f32_16x16x4_f32(false, a, false, b, (short)0, acc,
                                                false, false);
  }
  const int bi = t >> 12;
  const int l  = t & (LL - 1);
  #pragma unroll
  for (int i = 0; i < 8; ++i) {
    int o   = m0 + i + 8 * half;
    float v = acc[i];
    if (o < DE) xh_raw[((size_t)bi * DE + o) * LL + l] = v;                // x branch, planar
    else        z_gelu[(size_t)t * DE + (o - DE)]      = gelu_tanh(v);     // z branch, token-major
  }
}

// ---------------- Kernel 2: depthwise 3x3 conv (SAME) + bias + gelu ---------------
__global__ __launch_bounds__(256)
void k_dwconv(const float* __restrict__ xh_raw, const float* __restrict__ cw,
              const float* __restrict__ cb, float* __restrict__ xh) {
  int idx = blockIdx.x * blockDim.x + threadIdx.x;
  if (idx >= BATCH * DE * LL) return;
  int w = idx & (WW - 1);
  int h = (idx >> 6) & (HH - 1);
  int d = (idx >> 12) % DE;
  const float* plane = xh_raw + (size_t)(idx >> 12) * LL;
  const float* kw    = cw + d * 9;
  float s = 0.f;
  #pragma unroll
  for (int i = 0; i < 3; ++i) {
    int hh = h + i - 1;
    if (hh < 0 || hh >= HH) continue;
    #pragma unroll
    for (int j = 0; j < 3; ++j) {
      int wc = w + j - 1;
      if (wc < 0 || wc >= WW) continue;
      s += plane[hh * WW + wc] * kw[i * 3 + j];
    }
  }
  xh[idx] = gelu_tanh(s + cb[d]);
}

// ---------------- Kernel 3: x_proj (38x192 GEMM per direction, fused CrossScan) ---
__global__ __launch_bounds__(32)
void k_xproj(const float* __restrict__ xh, const float* __restrict__ xpw,
             float* __restrict__ xdbl) {
  const int lane = threadIdx.x;
  const int half = lane >> 4;
  const int lr   = lane & 15;
  const int m0   = blockIdx.x * 16;                    // row tile over 38 (padded 48)
  const int gn   = blockIdx.y * 16 + lr;               // over (b,k,l)
  const int l    = gn & (LL - 1);
  const int k    = (gn >> 12) & (KD - 1);
  const int b    = gn >> 14;
  const int mA   = m0 + lr;
  const int mAc  = (mA < CDL) ? mA : 0;                // clamp OOB reads
  const float* Abase = xpw + (size_t)(k * CDL + mAc) * DE;
  const float* Bbase = xh + (size_t)b * DE * LL + scan_pos(k, l);

  v8f acc = {};
  for (int d0 = 0; d0 < DE; d0 += 4) {
    v2f a, bb;
    int ks = d0 + 2 * half;
    a.x  = Abase[ks];
    a.y  = Abase[ks + 1];
    bb.x = Bbase[(size_t)ks * LL];          // CrossScan gather on the fly
    bb.y = Bbase[(size_t)(ks + 1) * LL];
    acc = __builtin_amdgcn_wmma_f32_16x16x4_f32(false, a, false, bb, (short)0, acc,
                                                false, false);
  }
  #pragma unroll
  for (int i = 0; i < 8; ++i) {
    int c = m0 + i + 8 * half;
    if (c < CDL)
      xdbl[(((size_t)b * KD + k) * CDL + c) * LL + l] = acc[i];
  }
}

// ---------------- Kernel 4: selective scan, fused dt-proj/softplus/merge ----------
// 16 lanes per (b,k,d) sequence; lane = state index n. y reduced via shfl_xor,
// scattered through scan_pos into the merged buffer with fp32 atomics.
__global__ __launch_bounds__(256)
void k_scan(const float* __restrict__ xh, const float* __restrict__ xdbl,
            const float* __restrict__ dtw, const float* __restrict__ dtb,
            const float* __restrict__ alogs, const float* __restrict__ Ds,
            float* __restrict__ ym) {
  int tid = blockIdx.x * blockDim.x + threadIdx.x;
  int seq = tid >> 4;
  int n   = tid & 15;
  if (seq >= BATCH * KD * DE) return;
  int d = seq % DE;
  int k = (seq / DE) % KD;
  int b = seq / (DE * KD);
  int kd = k * DE + d;

  float An  = -expf(alogs[kd * NS + n]);
  float w0 = dtw[kd * RR + 0], w1 = dtw[kd * RR + 1], w2 = dtw[kd * RR + 2];
  float w3 = dtw[kd * RR + 3], w4 = dtw[kd * RR + 4], w5 = dtw[kd * RR + 5];
  float bias = dtb[kd];
  float Dv   = Ds[kd];

  const float* xplane = xh + ((size_t)b * DE + d) * LL;
  const float* cbase  = xdbl + ((size_t)(b * KD + k) * CDL) * LL;
  float* yplane       = ym + ((size_t)b * DE + d) * LL;

  float h = 0.f;
  for (int l = 0; l < LL; ++l) {
    int pos = scan_pos(k, l);
    float u = xplane[pos];
    float s = bias + w0 * cbase[0 * LL + l] + w1 * cbase[1 * LL + l]
                   + w2 * cbase[2 * LL + l] + w3 * cbase[3 * LL + l]
                   + w4 * cbase[4 * LL + l] + w5 * cbase[5 * LL + l];
    float delta = (s > 20.f) ? s : log1pf(expf(s));        // softplus
    float Bn = cbase[(RR + n) * LL + l];
    float Cn = cbase[(RR + NS + n) * LL + l];
    h = expf(delta * An) * h + (delta * u) * Bn;
    float yp = h * Cn;
    yp += __shfl_xor(yp, 8);
    yp += __shfl_xor(yp, 4);
    yp += __shfl_xor(yp, 2);
    yp += __shfl_xor(yp, 1);
    if (n == 0) atomicAdd(&yplane[pos], yp + Dv * u);      // CrossMerge scatter
  }
}

// ---------------- Kernel 5: LayerNorm over D=192 + gate with gelu(z) --------------
__global__ __launch_bounds__(256)
void k_lngate(const float* __restrict__ ym, const float* __restrict__ zg,
              const float* __restrict__ lw, const float* __restrict__ lb,
              float* __restrict__ yln) {
  int lane  = threadIdx.x & 31;
  int token = blockIdx.x * (blockDim.x >> 5) + (threadIdx.x >> 5);
  int b = token >> 12;
  int l = token & (LL - 1);
  const float* ycol = ym + (size_t)b * DE * LL + l;
  float v[6];
  float s = 0.f, s2 = 0.f;
  #pragma unroll
  for (int i = 0; i < 6; ++i) {
    int d = lane + 32 * i;
    float t = ycol[(size_t)d * LL];
    v[i] = t; s += t; s2 += t * t;
  }
  #pragma unroll
  for (int off = 16; off > 0; off >>= 1) {
    s  += __shfl_xor(s, off);
    s2 += __shfl_xor(s2, off);
  }
  float mu  = s / DE;
  float var = s2 / DE - mu * mu;
  float inv = rsqrtf(var + LN_EPS);
  #pragma unroll
  for (int i = 0; i < 6; ++i) {
    int d = lane + 32 * i;
    float t = (v[i] - mu) * inv * lw[d] + lb[d];
    yln[(size_t)token * DE + d] = t * zg[(size_t)token * DE + d];
  }
}

// ---------------- Kernel 6: out_proj (96x192 GEMM) --------------------------------
__global__ __launch_bounds__(32)
void k_outproj(const float* __restrict__ yln, const float* __restrict__ wout,
               float* __restrict__ out) {
  const int lane = threadIdx.x;
  const int half = lane >> 4;
  const int lr   = lane & 15;
  const int m0   = blockIdx.x * 16;
  const int t    = blockIdx.y * 16 + lr;
  const int m    = m0 + lr;
  v8f acc = {};
  for (int d0 = 0; d0 < DE; d0 += 4) {
    v2f a, bb;
    int ks = d0 + 2 * half;
    a.x  = wout[m * DE + ks];
    a.y  = wout[m * DE + ks + 1];
    bb.x = yln[(size_t)t * DE + ks];
    bb.y = yln[(size_t)t * DE + ks + 1];
    acc = __builtin_amdgcn_wmma_f32_16x16x4_f32(false, a, false, bb, (short)0, acc,
                                                false, false);
  }
  #pragma unroll
  for (int i = 0; i < 8; ++i) {
    int c = m0 + i + 8 * half;
    out[(size_t)t * DM + c] = acc[i];
  }
}

extern "C" void kernel_launch(void* const* d_in, const int* in_sizes, int n_in,
                              void* d_out, int out_size, void* d_ws, size_t ws_size,
                              hipStream_t stream) {
  (void)in_sizes; (void)n_in; (void)out_size; (void)ws_size;
  const float* x    = (const float*)d_in[0];
  const float* win  = (const float*)d_in[1];
  const float* cw   = (const float*)d_in[2];
  const float* cb   = (const float*)d_in[3];
  const float* xpw  = (const float*)d_in[4];
  const float* dtw  = (const float*)d_in[5];
  const float* dtb  = (const float*)d_in[6];
  const float* alog = (const float*)d_in[7];
  const float* Ds   = (const float*)d_in[8];
  const float* lw   = (const float*)d_in[9];
  const float* lb   = (const float*)d_in[10];
  const float* wout = (const float*)d_in[11];
  float* out = (float*)d_out;

  float* ws = (float*)d_ws;
  const size_t S_PLANE = (size_t)BATCH * DE * LL;            // 3,145,728 floats
  float* xh_raw = ws;
  float* z_gelu = xh_raw + S_PLANE;
  float* xh     = z_gelu + S_PLANE;
  float* xdbl   = xh + S_PLANE;                              // (B,K,38,L)
  float* ym     = xdbl + (size_t)BATCH * KD * CDL * LL;
  float* yln    = ym + S_PLANE;

  k_inproj <<<dim3((2 * DE) / 16, (BATCH * LL) / 16), 32, 0, stream>>>(x, win, xh_raw, z_gelu);
  k_dwconv <<<(BATCH * DE * LL + 255) / 256, 256, 0, stream>>>(xh_raw, cw, cb, xh);
  k_xproj  <<<dim3(3, (BATCH * KD * LL) / 16), 32, 0, stream>>>(xh, xpw, xdbl);
  hipMemsetAsync(ym, 0, S_PLANE * sizeof(float), stream);
  k_scan   <<<(BATCH * KD * DE * 16) / 256, 256, 0, stream>>>(xh, xdbl, dtw, dtb, alog, Ds, ym);
  k_lngate <<<(BATCH * LL) / 8, 256, 0, stream>>>(ym, z_gelu, lw, lb, yln);
  k_outproj<<<dim3(DM / 16, (BATCH * LL) / 16), 32, 0, stream>>>(yln, wout, out);
}